// ConstantVelocityModel_68169720922999
// MI455X (gfx1250) — compile-verified
//
#include <hip/hip_runtime.h>
#include <math.h>

// ---------------------------------------------------------------------------
// CDNA5 / gfx1250 implementation of the constant-velocity latent-space model
// log-likelihood.  wave32, WGP, WMMA f32 reductions, TDM async LDS staging.
// ---------------------------------------------------------------------------

typedef __attribute__((ext_vector_type(2))) float v2f;
typedef __attribute__((ext_vector_type(8))) float v8f;

#define TPB 256          // 8 waves of 32 on CDNA5

// ---------------------------------------------------------------------------
// WMMA-based reduction of 256 per-thread partials (full f32 precision via
// V_WMMA_F32_16X16X4_F32 with a ones B-matrix).  Every wave executes the WMMA
// with EXEC all-ones (WMMA is not skipped on EXEC==0, so never branch around
// it); all waves compute the identical result, lane 0 returns it.
// ---------------------------------------------------------------------------
__device__ __forceinline__ float blockReduceWMMA(float val, float* sm) {
  sm[threadIdx.x] = val;
  __syncthreads();
  const int lane = (int)(threadIdx.x & 31u);
  float acc = 0.f;
#pragma unroll
  for (int k = 0; k < TPB / 32; ++k) acc += sm[lane + 32 * k];

  v2f A; A[0] = acc; A[1] = 0.f;   // A[m][1], A[m][3] = 0
  v2f B; B[0] = 1.f; B[1] = 1.f;   // 4x16 all-ones
  v8f C = {};
  v8f D = __builtin_amdgcn_wmma_f32_16x16x4_f32(
      /*neg_a=*/false, A, /*neg_b=*/false, B,
      /*c_mod=*/(short)0, C, /*reuse_a=*/false, /*reuse_b=*/false);

  float r = 0.f;
#pragma unroll
  for (int k = 0; k < 8; ++k) r += D[k];
  r += __shfl_down(r, 16, 32);     // lane0 += lane16
  return r;                        // valid in lane 0 (identical in all waves)
}

// ---------------------------------------------------------------------------
// Tensor Data Mover: async 1-D load of nElems f32 from global into LDS,
// optionally padding 2 DWORDs after every 2 DWORDs (interleaves two planes
// into a float4-stride-16B LDS layout).  Issued once per wave; TENSORcnt.
// ---------------------------------------------------------------------------
#if __has_builtin(__builtin_amdgcn_tensor_load_to_lds)
#define HAVE_TDM 1
typedef unsigned int v4u __attribute__((ext_vector_type(4)));
typedef int v8i __attribute__((ext_vector_type(8)));
typedef int v4i __attribute__((ext_vector_type(4)));

__device__ __forceinline__ void tdm_load_f32(const void* gsrc, unsigned ldsByteOff,
                                             unsigned nElems, bool interleavePad) {
  const unsigned long long ga = (unsigned long long)gsrc;
  v4u g0;
  g0[0] = 1u;                                        // count=1, user mode
  g0[1] = ldsByteOff;                                // lds_addr (bytes)
  g0[2] = (unsigned)(ga & 0xFFFFFFFFull);            // global_addr[31:0]
  g0[3] = (unsigned)((ga >> 32) & 0x1FFFFFFull)      // global_addr[56:32]
          | (2u << 30);                              // type = 2 ("image")
  unsigned w0 = (2u << 16);                          // data_size = 4 bytes
  if (interleavePad)
    w0 |= (1u << 20)                                 // pad_enable
        | (0u << 22)                                 // pad_interval: every 2 DW
        | (1u << 25);                                // pad_amount: 2 DW
  v8i g1;
  g1[0] = (int)w0;
  g1[1] = (int)((nElems & 0xFFFFu) << 16);           // tensor_dim0[15:0]
  g1[2] = (int)(((nElems >> 16) & 0xFFFFu)           // tensor_dim0[31:16]
                | (1u << 16));                       // tensor_dim1 = 1
  g1[3] = (int)((nElems & 0xFFFFu) << 16);           // tile_dim0 (<= 65535)
  g1[4] = 1;                                         // tile_dim1=1, tile_dim2=0
  g1[5] = (int)nElems;                               // tensor_dim0_stride lo
  g1[6] = 0;
  g1[7] = 0;
  const v4i z4 = (v4i)0;                             // groups 2/3 unused (<=2D)
#if defined(__clang_major__) && __clang_major__ >= 23
  const v8i z8 = (v8i)0;
  __builtin_amdgcn_tensor_load_to_lds(g0, g1, z4, z4, z8, 0);
#else
  __builtin_amdgcn_tensor_load_to_lds(g0, g1, z4, z4, 0);
#endif
}
#endif

// ---------------------------------------------------------------------------
// Kernel 1: event intensity sum  d_e = || (z_i - z_j) + (v_i - v_j) * t_e ||²
// Point table packed as float4 {zx, zy, vx, vy} in LDS -> 2 x ds_load_b128
// per event gather.  Staged via TDM (async tensor) and then rewritten by a
// plain coalesced loop so the layout is guaranteed regardless of D# padding
// semantics (same bytes when TDM behaves as documented).
// ---------------------------------------------------------------------------
__global__ __launch_bounds__(TPB) void cvm_events(
    const float2* __restrict__ z2, const float2* __restrict__ v2,
    const int2* __restrict__ eidx, const float* __restrict__ et,
    int nEvents, int n, float* __restrict__ partial) {
  __shared__ float4 pts[4096 + 2];   // +2: slack for the +8B v-plane TDM load
  __shared__ float red[TPB];

#if HAVE_TDM
  if (threadIdx.x < 32) {            // wave 0 drives the DMA (EXEC ignored)
    const unsigned base = (unsigned)(unsigned long long)(void*)&pts[0];
    tdm_load_f32(z2, base + 0u, 2u * (unsigned)n, true);   // -> .x/.y slots
    tdm_load_f32(v2, base + 8u, 2u * (unsigned)n, true);   // -> .z/.w slots
    __builtin_amdgcn_s_wait_tensorcnt(0);
  }
#endif
  __syncthreads();
  // Canonical staging (overwrites TDM result with identical data).
  for (int i = (int)threadIdx.x; i < n; i += TPB) {
    const float2 zi = z2[i];
    const float2 vi = v2[i];
    pts[i] = make_float4(zi.x, zi.y, vi.x, vi.y);
  }
  __syncthreads();

  float acc = 0.f;
  const int stride = (int)(gridDim.x * blockDim.x);
  for (int e = (int)(blockIdx.x * blockDim.x + threadIdx.x); e < nEvents; e += stride) {
    const int2 ij = eidx[e];
    const float t = et[e];
    const float4 pi = pts[ij.x];     // ds_load_b128
    const float4 pj = pts[ij.y];     // ds_load_b128
    const float dx = fmaf(pi.z - pj.z, t, pi.x - pj.x);
    const float dy = fmaf(pi.w - pj.w, t, pi.y - pj.y);
    acc = fmaf(dx, dx, acc);
    acc = fmaf(dy, dy, acc);
  }

  const float tot = blockReduceWMMA(acc, red);
  if (threadIdx.x == 0) partial[blockIdx.x] = tot;
}

// ---------------------------------------------------------------------------
// Kernel 2: non-event intensity over all i<j pairs (Gaussian erf integral).
// Block b processes rows {b, n-2-b} -> balanced ~n pairs per block.
// expo simplification: ((-b²+β)m² + 2abmn - n²(a²-β))/s == β - (bm - an)²/s
// ---------------------------------------------------------------------------
__global__ __launch_bounds__(TPB) void cvm_pairs(
    const float2* __restrict__ z, const float2* __restrict__ v,
    const float* __restrict__ betap, const float* __restrict__ t0p,
    const float* __restrict__ tnp, int n, float* __restrict__ partial) {
  __shared__ float red[TPB];

  const float beta = betap[0];
  const float t0   = t0p[0];
  const float tn   = tnp[0];
  const float HALF_SQRT_PI = 0.886226925452758014f;  // sqrt(pi)/2

  float acc = 0.f;
  const int b  = (int)blockIdx.x;
  const int i0 = b;
  const int i1 = n - 2 - b;

  for (int rr = 0; rr < 2; ++rr) {               // uniform across the block
    const int i = (rr == 0) ? i0 : i1;
    if (rr == 1 && i1 <= i0) break;
    const float2 zi = z[i];
    const float2 vi = v[i];
    for (int j = i + 1 + (int)threadIdx.x; j < n; j += TPB) {
      const float2 zj = z[j];
      const float2 vj = v[j];
      const float a_ = zi.x - zj.x;
      const float bb = zi.y - zj.y;
      const float m_ = vi.x - vj.x;
      const float nn = vi.y - vj.y;
      const float s  = m_ * m_ + nn * nn;
      const float sq = sqrtf(s);
      const float rs = 1.f / s;
      const float w  = bb * m_ - a_ * nn;        // cross term
      const float expo = beta - w * w * rs;
      const float amb = a_ * m_ + bb * nn;
      const float rsq = 1.f / sq;
      const float e0 = erff((s * t0 + amb) * rsq);
      const float e1 = erff((s * tn + amb) * rsq);
      acc += -HALF_SQRT_PI * expf(expo) * (e0 - e1) * rsq;
    }
  }

  const float tot = blockReduceWMMA(acc, red);
  if (threadIdx.x == 0) partial[blockIdx.x] = tot;
}

// ---------------------------------------------------------------------------
// Kernel 3: combine partials: ll = N_events*beta - (sum_d + sum_integ)
// ---------------------------------------------------------------------------
__global__ __launch_bounds__(TPB) void cvm_final(
    const float* __restrict__ wsA, int nA,
    const float* __restrict__ wsB, int nB,
    const float* __restrict__ betap, float nEventsF,
    float* __restrict__ out, int outSize) {
  __shared__ float red[TPB];
  float acc = 0.f;
  for (int k = (int)threadIdx.x; k < nA; k += TPB) acc += wsA[k];
  for (int k = (int)threadIdx.x; k < nB; k += TPB) acc += wsB[k];
  const float tot = blockReduceWMMA(acc, red);
  if (threadIdx.x == 0) {
    const float ll = nEventsF * betap[0] - tot;
    for (int k = 0; k < outSize; ++k) out[k] = ll;
  }
}

// ---------------------------------------------------------------------------
extern "C" void kernel_launch(void* const* d_in, const int* in_sizes, int n_in,
                              void* d_out, int out_size, void* d_ws, size_t ws_size,
                              hipStream_t stream) {
  const float2* z0  = (const float2*)d_in[0];
  const float2* v0  = (const float2*)d_in[1];
  const float* beta = (const float*)d_in[2];
  const int2*  eidx = (const int2*)d_in[3];
  const float* et   = (const float*)d_in[4];
  const float* t0   = (const float*)d_in[5];
  const float* tn   = (const float*)d_in[6];

  const int n       = in_sizes[0] / 2;   // 4096 points (x,y)
  const int nEvents = in_sizes[4];       // 2,000,000

  const int BLOCKS_A = 256;
  const int blocksB  = n / 2;            // 2048 balanced pair blocks

  float* wsA = (float*)d_ws;             // 256 partials (fully rewritten)
  float* wsB = wsA + BLOCKS_A;           // 2048 partials (fully rewritten)

  cvm_events<<<BLOCKS_A, TPB, 0, stream>>>(z0, v0, eidx, et, nEvents, n, wsA);
  cvm_pairs<<<blocksB, TPB, 0, stream>>>(z0, v0, beta, t0, tn, n, wsB);
  cvm_final<<<1, TPB, 0, stream>>>(wsA, BLOCKS_A, wsB, blocksB,
                                   beta, (float)nEvents, (float*)d_out, out_size);
}